// SFUnit_65850438582900
// MI455X (gfx1250) — compile-verified
//
#include <hip/hip_runtime.h>
#include <math.h>

#define DIMC   256
#define CONVC  256
#define PP     36
#define NTOK   144
#define BBATCH 8
#define HFC    56
#define TOK    (BBATCH*NTOK)   // 1152

typedef __attribute__((ext_vector_type(2))) float v2f;
typedef __attribute__((ext_vector_type(8))) float v8f;

__device__ __forceinline__ float gelu_exact(float x){
  return 0.5f * x * (1.0f + erff(x * 0.7071067811865476f));
}

__device__ __forceinline__ float block_sum256(float v, float* stmp){
  int tid = threadIdx.x;
  stmp[tid] = v; __syncthreads();
  #pragma unroll
  for (int s = 128; s > 0; s >>= 1){
    if (tid < s) stmp[tid] += stmp[tid + s];
    __syncthreads();
  }
  float r = stmp[0]; __syncthreads();
  return r;
}

// ---------------------------------------------------------------------------
// K1: per-token LayerNorm branches, ROI adjust, sampling-point generation,
//     mixing-vector h.  One block per token t = b*NTOK + n.
// ---------------------------------------------------------------------------
__global__ void __launch_bounds__(256)
k_prep(const float* __restrict__ emb, const float* __restrict__ roi_in,
       const float* __restrict__ adj_ln_w, const float* __restrict__ adj_ln_b,
       const float* __restrict__ adj_w,
       const float* __restrict__ off_ln_w, const float* __restrict__ off_ln_b,
       const float* __restrict__ off_w,   const float* __restrict__ off_bias,
       const float* __restrict__ pg_ln_w, const float* __restrict__ pg_ln_b,
       const float* __restrict__ pg1_w,   const float* __restrict__ pg1_b,
       float* __restrict__ pts_out, float* __restrict__ h_out,
       float* __restrict__ roi_out)
{
  int t = blockIdx.x;
  int b = t / NTOK, n = t % NTOK;
  int tid = threadIdx.x;
  __shared__ float stmp[256];
  __shared__ float sx[256];
  __shared__ float sOut[140];      // adj(4) | off(72) | h(64)
  __shared__ float sroi[4];
  __shared__ float sms[4];

  float x = emb[(size_t)(n*BBATCH + b)*DIMC + tid];
  float m   = block_sum256(x, stmp) * (1.0f/DIMC);
  float d   = x - m;
  float var = block_sum256(d*d, stmp) * (1.0f/DIMC);
  float rinv = rsqrtf(var + 1e-6f);
  sx[tid] = d * rinv;
  __syncthreads();

  if (tid < 140){
    const float* w; const float* lw; const float* lb; float bias;
    if (tid < 4)      { w = adj_w + tid*DIMC;        lw = adj_ln_w; lb = adj_ln_b; bias = 0.f; }
    else if (tid < 76){ int j = tid-4;  w = off_w + j*DIMC; lw = off_ln_w; lb = off_ln_b; bias = off_bias[j]; }
    else              { int j = tid-76; w = pg1_w + j*DIMC; lw = pg_ln_w;  lb = pg_ln_b;  bias = pg1_b[j]; }
    float acc = bias;
    for (int k = 0; k < DIMC; k++) acc += (sx[k]*lw[k] + lb[k]) * w[k];
    sOut[tid] = acc;
  }
  __syncthreads();

  if (tid == 0){
    const float* r = roi_in + (size_t)t*4;
    float cx = (r[2]+r[0])*0.5f, cy = (r[3]+r[1])*0.5f;
    float ww = fabsf(r[2]-r[0]), hh = fabsf(r[3]-r[1]);
    cx += sOut[0]*ww;  cy += sOut[1]*hh;
    ww *= expf(sOut[2]); hh *= expf(sOut[3]);
    float o0 = cx-0.5f*ww, o1 = cy-0.5f*hh, o2 = cx+0.5f*ww, o3 = cy+0.5f*hh;
    roi_out[(size_t)t*4+0] = o0; roi_out[(size_t)t*4+1] = o1;
    roi_out[(size_t)t*4+2] = o2; roi_out[(size_t)t*4+3] = o3;
    sroi[0]=o0; sroi[1]=o1; sroi[2]=o2; sroi[3]=o3;
  }
  if (tid < 2){                                 // per-coordinate mean / std(ddof=1)
    float mm = 0.f;
    for (int p = 0; p < PP; p++) mm += sOut[4 + 2*p + tid];
    mm *= (1.0f/PP);
    float vv = 0.f;
    for (int p = 0; p < PP; p++){ float dd = sOut[4+2*p+tid]-mm; vv += dd*dd; }
    sms[tid]   = mm;
    sms[2+tid] = sqrtf(vv * (1.0f/(PP-1))) + 1e-7f;
  }
  if (tid < 64) h_out[(size_t)t*64 + tid] = sOut[76+tid];
  __syncthreads();

  if (tid < 2*PP){
    int c = tid & 1;
    float off = sOut[4+tid];
    float on  = (off - sms[c]) / (3.0f * sms[2+c]);
    float center = (c==0) ? (sroi[0]+sroi[2])*0.5f : (sroi[1]+sroi[3])*0.5f;
    float ext    = (c==0) ? (sroi[2]-sroi[0])      : (sroi[3]-sroi[1]);
    pts_out[(size_t)t*PP*2 + tid] = center + on*ext;   // normalized [0,1]-ish coords
  }
}

// ---------------------------------------------------------------------------
// K2: bilinear grid-sample (zeros padding, align_corners=False).
//     One block per token, thread = channel, loop over 36 points.
// ---------------------------------------------------------------------------
__global__ void __launch_bounds__(256)
k_sample(const float* __restrict__ img, const float* __restrict__ pts,
         float* __restrict__ S)
{
  int t = blockIdx.x;
  int b = t / NTOK;
  int c = threadIdx.x;
  const float* imgb = img + ((size_t)b*CONVC + c) * (HFC*HFC);
  for (int p = 0; p < PP; p++){
    // grid = pts*2-1 ; x = ((g+1)*W - 1)*0.5  ==  pts*W - 0.5
    float px = pts[(size_t)t*PP*2 + p*2 + 0] * (float)HFC - 0.5f;
    float py = pts[(size_t)t*PP*2 + p*2 + 1] * (float)HFC - 0.5f;
    float x0f = floorf(px), y0f = floorf(py);
    int   x0 = (int)x0f,    y0 = (int)y0f;
    float wx = px - x0f,    wy = py - y0f;
    float v = 0.f;
    #pragma unroll
    for (int dy = 0; dy < 2; dy++){
      #pragma unroll
      for (int dx = 0; dx < 2; dx++){
        int xi = x0 + dx, yi = y0 + dy;
        float w = (dx ? wx : 1.f-wx) * (dy ? wy : 1.f-wy);
        if (xi >= 0 && xi < HFC && yi >= 0 && yi < HFC)
          v += w * imgb[yi*HFC + xi];
      }
    }
    S[(size_t)t*PP*CONVC + p*CONVC + c] = v;
  }
}

// ---------------------------------------------------------------------------
// K3: fused channel-mix-matrix generation + mix1 (WMMA fp32 16x16x4).
//     One block (8 waves) per token.  O1 = gelu(S @ CM + m_beta).
// ---------------------------------------------------------------------------
__global__ void __launch_bounds__(256)
k_mix1(const float* __restrict__ S, const float* __restrict__ h_in,
       const float* __restrict__ W2, const float* __restrict__ b2,
       const float* __restrict__ m_beta, float* __restrict__ O1)
{
  int t   = blockIdx.x;
  int tid = threadIdx.x;
  int wave = tid >> 5, lane = tid & 31;
  int half = lane >> 4, l16 = lane & 15;

  __shared__ float sCM[16][260];   // 16 c-rows x 256 d (padded)
  __shared__ float sS[48][17];     // 48 p-rows (pad >=36 zero) x 16 k
  __shared__ float sH[64];

  if (tid < 64) sH[tid] = h_in[(size_t)t*64 + tid];
  for (int i = tid; i < 48*17; i += 256) (&sS[0][0])[i] = 0.f;
  __syncthreads();

  float hreg[64];
  #pragma unroll
  for (int k = 0; k < 64; k++) hreg[k] = sH[k];

  v8f zero8 = {0.f,0.f,0.f,0.f,0.f,0.f,0.f,0.f};
  v8f acc[3][2];
  #pragma unroll
  for (int i = 0; i < 3; i++){ acc[i][0] = zero8; acc[i][1] = zero8; }

  int nt0 = wave * 2;
  const float* Stok = S + (size_t)t*PP*CONVC;

  for (int c0 = 0; c0 < CONVC; c0 += 16){
    // stage S[:, c0:c0+16]
    for (int i = tid; i < PP*16; i += 256){
      int p = i >> 4, kk = i & 15;
      sS[p][kk] = Stok[p*CONVC + c0 + kk];
    }
    // generate CM rows c0..c0+15 : CM[c][d] = h . W2[c*256+d] + b2[c*256+d]
    {
      int crow  = tid >> 4;
      int dbase = (tid & 15) * 16;
      int c = c0 + crow;
      const float* wrow = W2 + (size_t)(c*CONVC + dbase) * 64;
      const float* brow = b2 + (size_t)(c*CONVC + dbase);
      #pragma unroll
      for (int i = 0; i < 16; i++){
        float a = brow[i];
        const float* w = wrow + i*64;
        #pragma unroll
        for (int k = 0; k < 64; k++) a += hreg[k] * w[k];
        sCM[crow][dbase + i] = a;
      }
    }
    __syncthreads();
    #pragma unroll
    for (int kk = 0; kk < 16; kk += 4){
      int kA = kk + 2*half;
      v2f b0, b1;
      b0.x = sCM[kA  ][ nt0   *16 + l16];
      b0.y = sCM[kA+1][ nt0   *16 + l16];
      b1.x = sCM[kA  ][(nt0+1)*16 + l16];
      b1.y = sCM[kA+1][(nt0+1)*16 + l16];
      #pragma unroll
      for (int mt = 0; mt < 3; mt++){
        v2f a;
        a.x = sS[mt*16 + l16][kA];
        a.y = sS[mt*16 + l16][kA+1];
        acc[mt][0] = __builtin_amdgcn_wmma_f32_16x16x4_f32(false, a, false, b0, (short)0, acc[mt][0], false, false);
        acc[mt][1] = __builtin_amdgcn_wmma_f32_16x16x4_f32(false, a, false, b1, (short)0, acc[mt][1], false, false);
      }
    }
    __syncthreads();
  }

  float* Ot = O1 + (size_t)t*PP*CONVC;
  #pragma unroll
  for (int j = 0; j < 2; j++){
    int dcol = (nt0+j)*16 + l16;
    float mb = m_beta[dcol];
    #pragma unroll
    for (int mt = 0; mt < 3; mt++){
      #pragma unroll
      for (int r = 0; r < 8; r++){
        int prow = mt*16 + r + 8*half;
        if (prow < PP)
          Ot[prow*CONVC + dcol] = gelu_exact(acc[mt][j][r] + mb);
      }
    }
  }
}

// ---------------------------------------------------------------------------
// K4: fused spatial-mix-matrix generation + mix2 (WMMA fp32 16x16x4).
//     O2 = gelu(SM @ O1 + s_beta[row]).  One block per token.
// ---------------------------------------------------------------------------
__global__ void __launch_bounds__(256)
k_mix2(const float* __restrict__ O1, const float* __restrict__ h_in,
       const float* __restrict__ W2, const float* __restrict__ b2,
       const float* __restrict__ s_beta, float* __restrict__ O2)
{
  int t   = blockIdx.x;
  int tid = threadIdx.x;
  int wave = tid >> 5, lane = tid & 31;
  int half = lane >> 4, l16 = lane & 15;

  __shared__ float sSM[48][44];    // q-rows x p-cols (K padded to 40)
  __shared__ float sO1[40][260];   // p-rows (pad >=36 zero) x 256 d
  __shared__ float sH[64];

  if (tid < 64) sH[tid] = h_in[(size_t)t*64 + tid];
  for (int i = tid; i < 48*44; i += 256) (&sSM[0][0])[i] = 0.f;
  for (int i = tid; i < 4*260;  i += 256) (&sO1[36][0])[i] = 0.f;
  __syncthreads();

  float hreg[64];
  #pragma unroll
  for (int k = 0; k < 64; k++) hreg[k] = sH[k];

  // generate SM: prm row index R = 65536 + q*36 + p
  for (int i = tid; i < PP*PP; i += 256){
    int R = CONVC*CONVC + i;
    const float* w = W2 + (size_t)R*64;
    float a = b2[R];
    #pragma unroll
    for (int k = 0; k < 64; k++) a += hreg[k] * w[k];
    sSM[i/PP][i%PP] = a;
  }
  // load O1 tile
  const float* O1t = O1 + (size_t)t*PP*CONVC;
  for (int i = tid; i < PP*CONVC; i += 256)
    sO1[i >> 8][i & 255] = O1t[i];
  __syncthreads();

  v8f zero8 = {0.f,0.f,0.f,0.f,0.f,0.f,0.f,0.f};
  v8f acc[3][2];
  #pragma unroll
  for (int i = 0; i < 3; i++){ acc[i][0] = zero8; acc[i][1] = zero8; }
  int nt0 = wave * 2;

  #pragma unroll
  for (int kk = 0; kk < 40; kk += 4){
    int kA = kk + 2*half;
    v2f b0, b1;
    b0.x = sO1[kA  ][ nt0   *16 + l16];
    b0.y = sO1[kA+1][ nt0   *16 + l16];
    b1.x = sO1[kA  ][(nt0+1)*16 + l16];
    b1.y = sO1[kA+1][(nt0+1)*16 + l16];
    #pragma unroll
    for (int mt = 0; mt < 3; mt++){
      v2f a;
      a.x = sSM[mt*16 + l16][kA];
      a.y = sSM[mt*16 + l16][kA+1];
      acc[mt][0] = __builtin_amdgcn_wmma_f32_16x16x4_f32(false, a, false, b0, (short)0, acc[mt][0], false, false);
      acc[mt][1] = __builtin_amdgcn_wmma_f32_16x16x4_f32(false, a, false, b1, (short)0, acc[mt][1], false, false);
    }
  }

  float* O2t = O2 + (size_t)t*PP*CONVC;
  #pragma unroll
  for (int j = 0; j < 2; j++){
    int d = (nt0+j)*16 + l16;
    #pragma unroll
    for (int mt = 0; mt < 3; mt++){
      #pragma unroll
      for (int r = 0; r < 8; r++){
        int q = mt*16 + r + 8*half;
        if (q < PP)
          O2t[q*CONVC + d] = gelu_exact(acc[mt][j][r] + s_beta[q]);
      }
    }
  }
}

// ---------------------------------------------------------------------------
// K5: output projection GEMM (M=1152 tokens, N=256, K=9216) + residual.
//     One block per 16 tokens.  xmid(N,B,D) = emb + mixed.
// ---------------------------------------------------------------------------
__global__ void __launch_bounds__(256)
k_opproj(const float* __restrict__ O2, const float* __restrict__ op_w,
         const float* __restrict__ op_b, const float* __restrict__ emb,
         float* __restrict__ xmid)
{
  int tok0 = blockIdx.x * 16;
  int tid  = threadIdx.x;
  int wave = tid >> 5, lane = tid & 31;
  int half = lane >> 4, l16 = lane & 15;

  __shared__ float sA[16][20];     // tokens x 16 k
  __shared__ float sB[16][260];    // 16 k x 256 n

  v8f zero8 = {0.f,0.f,0.f,0.f,0.f,0.f,0.f,0.f};
  v8f acc[2]; acc[0] = zero8; acc[1] = zero8;
  int nt0 = wave * 2;

  for (int k0 = 0; k0 < PP*CONVC; k0 += 16){
    {
      int mm = tid >> 4, kk = tid & 15;
      sA[mm][kk] = O2[(size_t)(tok0 + mm)*PP*CONVC + k0 + kk];
    }
    {
      const float* w = op_w + (size_t)tid * (PP*CONVC) + k0;
      #pragma unroll
      for (int kk = 0; kk < 16; kk++) sB[kk][tid] = w[kk];
    }
    __syncthreads();
    #pragma unroll
    for (int kk = 0; kk < 16; kk += 4){
      int kA = kk + 2*half;
      v2f a;  a.x = sA[l16][kA];  a.y = sA[l16][kA+1];
      v2f b0, b1;
      b0.x = sB[kA  ][ nt0   *16 + l16];
      b0.y = sB[kA+1][ nt0   *16 + l16];
      b1.x = sB[kA  ][(nt0+1)*16 + l16];
      b1.y = sB[kA+1][(nt0+1)*16 + l16];
      acc[0] = __builtin_amdgcn_wmma_f32_16x16x4_f32(false, a, false, b0, (short)0, acc[0], false, false);
      acc[1] = __builtin_amdgcn_wmma_f32_16x16x4_f32(false, a, false, b1, (short)0, acc[1], false, false);
    }
    __syncthreads();
  }

  #pragma unroll
  for (int j = 0; j < 2; j++){
    int nd = (nt0+j)*16 + l16;
    float ob = op_b[nd];
    #pragma unroll
    for (int r = 0; r < 8; r++){
      int m  = r + 8*half;
      int tok = tok0 + m;                 // tok = b*NTOK + n
      int bb = tok / NTOK, nn = tok % NTOK;
      size_t xi = (size_t)(nn*BBATCH + bb)*DIMC + nd;
      xmid[xi] = emb[xi] + acc[j][r] + ob;
    }
  }
}

// ---------------------------------------------------------------------------
// K6a: attention LN + QKV projection.  One block per (n,b) row s = n*B+b.
// ---------------------------------------------------------------------------
__global__ void __launch_bounds__(256)
k_qkv(const float* __restrict__ xmid, const float* __restrict__ ln_w,
      const float* __restrict__ ln_b, const float* __restrict__ ipw,
      const float* __restrict__ ipb, float* __restrict__ qkv)
{
  int s = blockIdx.x;
  int tid = threadIdx.x;
  __shared__ float stmp[256];
  __shared__ float sa[256];
  float x = xmid[(size_t)s*DIMC + tid];
  float m   = block_sum256(x, stmp) * (1.f/DIMC);
  float d   = x - m;
  float var = block_sum256(d*d, stmp) * (1.f/DIMC);
  float rinv = rsqrtf(var + 1e-6f);
  sa[tid] = d*rinv*ln_w[tid] + ln_b[tid];
  __syncthreads();
  #pragma unroll
  for (int jj = 0; jj < 3; jj++){
    int j = tid + 256*jj;
    const float* w = ipw + (size_t)j*DIMC;
    float a = ipb[j];
    for (int k = 0; k < DIMC; k++) a += sa[k]*w[k];
    qkv[(size_t)s*768 + j] = a;
  }
}

// ---------------------------------------------------------------------------
// K6b: 64-head attention, head_dim=4, online softmax. One block per (b,h).
// ---------------------------------------------------------------------------
__global__ void __launch_bounds__(256)
k_attn(const float* __restrict__ qkv, float* __restrict__ ao)
{
  int bh = blockIdx.x;
  int b = bh >> 6, hh = bh & 63;
  int tid = threadIdx.x;
  __shared__ float sQ[NTOK][4], sK[NTOK][4], sV[NTOK][4];
  for (int i = tid; i < NTOK*4; i += 256){
    int n = i >> 2, d = i & 3;
    size_t base = (size_t)(n*BBATCH + b)*768;
    sQ[n][d] = qkv[base +        hh*4 + d];
    sK[n][d] = qkv[base + 256 +  hh*4 + d];
    sV[n][d] = qkv[base + 512 +  hh*4 + d];
  }
  __syncthreads();
  if (tid < NTOK){
    float q0=sQ[tid][0], q1=sQ[tid][1], q2=sQ[tid][2], q3=sQ[tid][3];
    float mx = -3.4e38f, l = 0.f, a0=0.f, a1=0.f, a2=0.f, a3=0.f;
    for (int k = 0; k < NTOK; k++){
      float sc = (q0*sK[k][0] + q1*sK[k][1] + q2*sK[k][2] + q3*sK[k][3]) * 0.5f;
      float nm = fmaxf(mx, sc);
      float corr = expf(mx - nm);
      float p = expf(sc - nm);
      l  = l*corr + p;
      a0 = a0*corr + p*sV[k][0];
      a1 = a1*corr + p*sV[k][1];
      a2 = a2*corr + p*sV[k][2];
      a3 = a3*corr + p*sV[k][3];
      mx = nm;
    }
    float rl = 1.f / l;
    size_t base = (size_t)(tid*BBATCH + b)*DIMC + hh*4;
    ao[base+0]=a0*rl; ao[base+1]=a1*rl; ao[base+2]=a2*rl; ao[base+3]=a3*rl;
  }
}

// ---------------------------------------------------------------------------
// K6c: attention output projection + residual.
// ---------------------------------------------------------------------------
__global__ void __launch_bounds__(256)
k_attnout(const float* __restrict__ ao, const float* __restrict__ ow,
          const float* __restrict__ ob, const float* __restrict__ xmid,
          float* __restrict__ x2)
{
  int s = blockIdx.x;
  int tid = threadIdx.x;
  __shared__ float sa[256];
  sa[tid] = ao[(size_t)s*DIMC + tid];
  __syncthreads();
  const float* w = ow + (size_t)tid*DIMC;
  float a = ob[tid];
  for (int k = 0; k < DIMC; k++) a += sa[k]*w[k];
  x2[(size_t)s*DIMC + tid] = xmid[(size_t)s*DIMC + tid] + a;
}

// ---------------------------------------------------------------------------
// K7: FFN (LN -> 1024 gelu -> 256) + residual; writes final x to d_out.
// ---------------------------------------------------------------------------
__global__ void __launch_bounds__(256)
k_ffn(const float* __restrict__ x2, const float* __restrict__ ln_w,
      const float* __restrict__ ln_b, const float* __restrict__ w1,
      const float* __restrict__ b1, const float* __restrict__ w2,
      const float* __restrict__ b2v, float* __restrict__ out)
{
  int s = blockIdx.x;
  int tid = threadIdx.x;
  __shared__ float stmp[256];
  __shared__ float sf[256];
  __shared__ float sh1[1024];
  float x = x2[(size_t)s*DIMC + tid];
  float m   = block_sum256(x, stmp) * (1.f/DIMC);
  float d   = x - m;
  float var = block_sum256(d*d, stmp) * (1.f/DIMC);
  float rinv = rsqrtf(var + 1e-6f);
  sf[tid] = d*rinv*ln_w[tid] + ln_b[tid];
  __syncthreads();
  #pragma unroll
  for (int jj = 0; jj < 4; jj++){
    int j = tid + 256*jj;
    const float* w = w1 + (size_t)j*DIMC;
    float a = b1[j];
    for (int k = 0; k < DIMC; k++) a += sf[k]*w[k];
    sh1[j] = gelu_exact(a);
  }
  __syncthreads();
  const float* w = w2 + (size_t)tid*1024;
  float a = b2v[tid];
  for (int k = 0; k < 1024; k++) a += sh1[k]*w[k];
  out[(size_t)s*DIMC + tid] = x + a;
}

// ---------------------------------------------------------------------------
extern "C" void kernel_launch(void* const* d_in, const int* in_sizes, int n_in,
                              void* d_out, int out_size, void* d_ws, size_t ws_size,
                              hipStream_t stream)
{
  (void)in_sizes; (void)n_in; (void)out_size; (void)ws_size;
  const float* emb      = (const float*)d_in[0];
  const float* roi_in   = (const float*)d_in[1];
  const float* img      = (const float*)d_in[2];
  const float* adj_ln_w = (const float*)d_in[3];
  const float* adj_ln_b = (const float*)d_in[4];
  const float* adj_w    = (const float*)d_in[5];
  const float* off_ln_w = (const float*)d_in[6];
  const float* off_ln_b = (const float*)d_in[7];
  const float* off_w    = (const float*)d_in[8];
  const float* off_bias = (const float*)d_in[9];
  const float* pg_ln_w  = (const float*)d_in[10];
  const float* pg_ln_b  = (const float*)d_in[11];
  const float* pg1_w    = (const float*)d_in[12];
  const float* pg1_b    = (const float*)d_in[13];
  const float* pg2_w    = (const float*)d_in[14];
  const float* pg2_b    = (const float*)d_in[15];
  const float* m_beta   = (const float*)d_in[16];
  const float* s_beta   = (const float*)d_in[17];
  const float* op_w     = (const float*)d_in[18];
  const float* op_b     = (const float*)d_in[19];
  const float* attn_ipw = (const float*)d_in[20];
  const float* attn_ipb = (const float*)d_in[21];
  const float* attn_ow  = (const float*)d_in[22];
  const float* attn_ob  = (const float*)d_in[23];
  const float* lnattn_w = (const float*)d_in[24];
  const float* lnattn_b = (const float*)d_in[25];
  const float* ffn_ln_w = (const float*)d_in[26];
  const float* ffn_ln_b = (const float*)d_in[27];
  const float* ffn1_w   = (const float*)d_in[28];
  const float* ffn1_b   = (const float*)d_in[29];
  const float* ffn2_w   = (const float*)d_in[30];
  const float* ffn2_b   = (const float*)d_in[31];

  float* out = (float*)d_out;
  float* x_out   = out;                         // (N,B,D) 294912
  float* roi_out = out + (size_t)NTOK*BBATCH*DIMC;

  // workspace carve-up (floats)
  float* ws   = (float*)d_ws;
  float* pts  = ws;                              //  82944
  float* hbuf = ws + 82944;                      //  73728
  float* xmid = ws + 156672;                     // 294912
  float* x2b  = ws + 451584;                     // 294912
  float* Sbuf = ws + 746496;                     // 10616832 (S; reused as O2)
  float* O1b  = ws + 11363328;                   // 10616832 (O1; reused qkv+ao)
  float* qkvb = O1b;                             //  884736
  float* aob  = O1b + 884736;                    //  294912

  k_prep<<<TOK, 256, 0, stream>>>(emb, roi_in, adj_ln_w, adj_ln_b, adj_w,
                                  off_ln_w, off_ln_b, off_w, off_bias,
                                  pg_ln_w, pg_ln_b, pg1_w, pg1_b,
                                  pts, hbuf, roi_out);
  k_sample<<<TOK, 256, 0, stream>>>(img, pts, Sbuf);
  k_mix1<<<TOK, 256, 0, stream>>>(Sbuf, hbuf, pg2_w, pg2_b, m_beta, O1b);
  k_mix2<<<TOK, 256, 0, stream>>>(O1b, hbuf, pg2_w, pg2_b, s_beta, Sbuf /*O2*/);
  k_opproj<<<TOK/16, 256, 0, stream>>>(Sbuf, op_w, op_b, emb, xmid);
  k_qkv<<<NTOK*BBATCH, 256, 0, stream>>>(xmid, lnattn_w, lnattn_b, attn_ipw, attn_ipb, qkvb);
  k_attn<<<BBATCH*64, 256, 0, stream>>>(qkvb, aob);
  k_attnout<<<NTOK*BBATCH, 256, 0, stream>>>(aob, attn_ow, attn_ob, xmid, x2b);
  k_ffn<<<NTOK*BBATCH, 256, 0, stream>>>(x2b, ffn_ln_w, ffn_ln_b,
                                         ffn1_w, ffn1_b, ffn2_w, ffn2_b, x_out);
}